// BinaryModule_71012989272077
// MI455X (gfx1250) — compile-verified
//
#include <hip/hip_runtime.h>

typedef __attribute__((ext_vector_type(2))) float v2f;
typedef __attribute__((ext_vector_type(8))) float v8f;

#define COLS            8192
#define ROWS_PER_BLOCK  16
#define THREADS         256
#define MASK_DWORDS     16   // 128 float4 per thread * 4 bits = 512 bits

__global__ __launch_bounds__(THREADS)
void binarize_rowalpha_kernel(const float* __restrict__ x, float* __restrict__ out) {
  __shared__ unsigned smask[MASK_DWORDS * THREADS];   // 16 KB sign-bit cache
  __shared__ float sP[ROWS_PER_BLOCK * 17];           // per-thread |x| partials (padded)
  __shared__ float cP[ROWS_PER_BLOCK * 17];           // per-thread count partials
  __shared__ float sRow[ROWS_PER_BLOCK];
  __shared__ float cRow[ROWS_PER_BLOCK];

  const int t = threadIdx.x;
  const int r = t >> 4;          // row within block: 0..15
  const int j = t & 15;          // thread within row: 0..15
  const long long row = (long long)blockIdx.x * ROWS_PER_BLOCK + r;
  const float* xrow = x + row * (long long)COLS;

  // ---- Stage 1: stream row once; abs-sum, nonzero count, pack sign bits ----
  float s = 0.f, c = 0.f;
  for (int d = 0; d < MASK_DWORDS; ++d) {
    unsigned mask = 0u;
#pragma unroll
    for (int i = 0; i < 8; ++i) {
      const int col = (d * 8 + i) * 64 + j * 4;       // coalesced b128 loads
      const float4 v = *(const float4*)(xrow + col);
      const float ax = fabsf(v.x), ay = fabsf(v.y), az = fabsf(v.z), aw = fabsf(v.w);
      s += ax + ay + az + aw;
      c += (ax > 0.f ? 1.f : 0.f) + (ay > 0.f ? 1.f : 0.f)
         + (az > 0.f ? 1.f : 0.f) + (aw > 0.f ? 1.f : 0.f);
      mask |= (v.x > 0.f ? 1u : 0u) << (i * 4 + 0);
      mask |= (v.y > 0.f ? 1u : 0u) << (i * 4 + 1);
      mask |= (v.z > 0.f ? 1u : 0u) << (i * 4 + 2);
      mask |= (v.w > 0.f ? 1u : 0u) << (i * 4 + 3);
    }
    smask[d * THREADS + t] = mask;                    // conflict-free: stride 1 per d
  }

  sP[r * 17 + j] = s;
  cP[r * 17 + j] = c;
  __syncthreads();

  // ---- Stage 2: 16x16 row-sum via V_WMMA_F32_16X16X4_F32 against all-ones B ----
  // A = 16x4 chunk of partials (lane<16: K=kb,kb+1; lane>=16: K=kb+2,kb+3),
  // B = 4x16 ones => D[m][n] = rowsum(A[m][*]) for every n; accumulate 4 chunks.
  // All 8 waves compute identically (EXEC all-1s, no divergence around WMMA);
  // duplicate identical LDS writes are benign.
  {
    const int lane  = t & 31;
    const int m     = lane & 15;
    const int khalf = (lane >> 4) << 1;               // 0 for lanes 0-15, 2 for 16-31
    v8f accS = {0.f, 0.f, 0.f, 0.f, 0.f, 0.f, 0.f, 0.f};
    v8f accC = {0.f, 0.f, 0.f, 0.f, 0.f, 0.f, 0.f, 0.f};
    const v2f ones = {1.f, 1.f};
#pragma unroll
    for (int q = 0; q < 4; ++q) {
      const int kb = 4 * q + khalf;
      v2f As = { sP[m * 17 + kb], sP[m * 17 + kb + 1] };
      v2f Ac = { cP[m * 17 + kb], cP[m * 17 + kb + 1] };
      accS = __builtin_amdgcn_wmma_f32_16x16x4_f32(false, As, false, ones,
                                                   (short)0, accS, false, false);
      accC = __builtin_amdgcn_wmma_f32_16x16x4_f32(false, Ac, false, ones,
                                                   (short)0, accC, false, false);
    }
    // D layout: VGPR v = row v (lanes 0-15) / row v+8 (lanes 16-31); N = lane%16.
    if ((lane & 15) == 0) {                           // lanes 0 and 16 publish
      const int mbase = (lane >> 4) * 8;
#pragma unroll
      for (int v = 0; v < 8; ++v) {
        sRow[mbase + v] = accS[v];
        cRow[mbase + v] = accC[v];
      }
    }
  }
  __syncthreads();

  // ---- Stage 3: alpha = clip(abssum/count, 0, 100); emit +/-alpha from masks ----
  const float alpha  = fminf(fmaxf(sRow[r] / cRow[r], 0.f), 100.f);
  const float nalpha = -alpha;
  float* orow = out + row * (long long)COLS;
  for (int d = 0; d < MASK_DWORDS; ++d) {
    const unsigned mask = smask[d * THREADS + t];
#pragma unroll
    for (int i = 0; i < 8; ++i) {
      const int col = (d * 8 + i) * 64 + j * 4;
      float4 o;
      o.x = ((mask >> (i * 4 + 0)) & 1u) ? alpha : nalpha;
      o.y = ((mask >> (i * 4 + 1)) & 1u) ? alpha : nalpha;
      o.z = ((mask >> (i * 4 + 2)) & 1u) ? alpha : nalpha;
      o.w = ((mask >> (i * 4 + 3)) & 1u) ? alpha : nalpha;
      *(float4*)(orow + col) = o;                     // coalesced b128 stores
    }
  }
}

extern "C" void kernel_launch(void* const* d_in, const int* in_sizes, int n_in,
                              void* d_out, int out_size, void* d_ws, size_t ws_size,
                              hipStream_t stream) {
  (void)n_in; (void)out_size; (void)d_ws; (void)ws_size;
  const float* x = (const float*)d_in[0];
  float* out = (float*)d_out;
  const int rows = in_sizes[0] / COLS;                // 16384
  dim3 grid(rows / ROWS_PER_BLOCK);                   // 1024 blocks
  binarize_rowalpha_kernel<<<grid, dim3(THREADS), 0, stream>>>(x, out);
}